// MSDeformAttn_32736240730931
// MI455X (gfx1250) — compile-verified
//
#include <hip/hip_runtime.h>
#include <hip/hip_bf16.h>
#include <math.h>

// ---------------- problem constants (fixed shapes) ----------------
#define C_DIM 256
#define NH    8
#define NL    4
#define NP    4
#define HD    32            // C_DIM / NH
#define B_SZ  2
#define LQ    21760
#define LIN   21760

typedef __attribute__((ext_vector_type(2))) float v2f;
typedef __attribute__((ext_vector_type(8))) float v8f;

// =================================================================
// GEMM:  C[M,N] = A[M,K] * W[K,N] + bias[N]   (all row-major fp32)
// M % 16 == 0, N % 64 == 0, K % 4 == 0.
// One wave32 computes a 16x64 output strip using V_WMMA_F32_16X16X4_F32.
// A-operand layout (ISA 7.12.2, 32-bit A 16x4): lane L holds
//   a = { A[m][k + 2*kh], A[m][k + 2*kh + 1] },  m = L%16, kh = L/16.
// B-operand (4x16, row striped across lanes): lane L holds
//   b = { W[k + 2*kh][n], W[k + 2*kh + 1][n] },  n = L%16.
// C/D layout: element i of v8f is row (i + 8*kh), col n.
// =================================================================
__global__ __launch_bounds__(256) void gemm_f32_wmma(
    const float* __restrict__ A, const float* __restrict__ W,
    const float* __restrict__ bias, float* __restrict__ Cm,
    int M, int N, int K) {
  const int lane = threadIdx.x & 31;
  const int wave = threadIdx.x >> 5;
  const int mt   = blockIdx.x * 8 + wave;   // 16-row tile index
  if (mt * 16 >= M) return;                 // wave-uniform guard (EXEC stays full)
  const int row0 = mt * 16;
  const int n0   = blockIdx.y * 64;

  const int m  = lane & 15;
  const int kh = lane >> 4;                 // 0 or 1

  v8f acc0 = {}; v8f acc1 = {}; v8f acc2 = {}; v8f acc3 = {};

  const float* __restrict__ Arow = A + (size_t)(row0 + m) * K;

  for (int k = 0; k < K; k += 4) {
    v2f a = *(const v2f*)(Arow + k + 2 * kh);
    const float* __restrict__ Wk0 = W + (size_t)(k + 2 * kh) * N + n0 + m;
    const float* __restrict__ Wk1 = Wk0 + N;
    v2f b0 = { Wk0[ 0], Wk1[ 0] };
    v2f b1 = { Wk0[16], Wk1[16] };
    v2f b2 = { Wk0[32], Wk1[32] };
    v2f b3 = { Wk0[48], Wk1[48] };
    acc0 = __builtin_amdgcn_wmma_f32_16x16x4_f32(false, a, false, b0, (short)0, acc0, false, false);
    acc1 = __builtin_amdgcn_wmma_f32_16x16x4_f32(false, a, false, b1, (short)0, acc1, false, false);
    acc2 = __builtin_amdgcn_wmma_f32_16x16x4_f32(false, a, false, b2, (short)0, acc2, false, false);
    acc3 = __builtin_amdgcn_wmma_f32_16x16x4_f32(false, a, false, b3, (short)0, acc3, false, false);
  }

  v8f accs[4] = { acc0, acc1, acc2, acc3 };
#pragma unroll
  for (int t = 0; t < 4; ++t) {
    const int ncol = n0 + t * 16 + m;
    const float bv = bias[ncol];
#pragma unroll
    for (int i = 0; i < 8; ++i) {
      const int row = row0 + i + 8 * kh;
      Cm[(size_t)row * N + ncol] = accs[t][i] + bv;
    }
  }
}

// =================================================================
// Fused softmax + multi-scale deformable sampling.
// One wave per (b, q, h); lane = channel d (HD == 32 == wave32).
//   value  : (B, LIN, NH, HD)
//   refpts : (B, LQ, NL, 2)
//   offs   : (B, LQ, NH, NL, NP, 2)  -> 32 scalars per (b,q,h)
//   logits : (B, LQ, NH, 16)
//   outpre : (B, LQ, NH, HD) == (B, LQ, C)
// =================================================================
__global__ __launch_bounds__(256) void msda_sample(
    const float* __restrict__ value, const float* __restrict__ refpts,
    const float* __restrict__ offs,  const float* __restrict__ logits,
    float* __restrict__ outpre) {
  const int lane = threadIdx.x & 31;
  int w = (int)((blockIdx.x * (unsigned)blockDim.x + threadIdx.x) >> 5);
  if (w >= B_SZ * LQ * NH) return;
  const int h = w % NH;  w /= NH;
  const int q = w % LQ;  w /= LQ;
  const int b = w;

  const size_t qh = ((size_t)b * LQ + q) * NH + h;

  // ---- softmax over 16 logits (lanes 0..15 active) ----
  float lg = (lane < 16) ? logits[qh * 16 + lane] : -3.402823466e38f;
  float mx = lg;
  mx = fmaxf(mx, __shfl_xor(mx, 8));
  mx = fmaxf(mx, __shfl_xor(mx, 4));
  mx = fmaxf(mx, __shfl_xor(mx, 2));
  mx = fmaxf(mx, __shfl_xor(mx, 1));
  float ex = (lane < 16) ? expf(lg - mx) : 0.0f;
  float sm = ex;
  sm += __shfl_xor(sm, 8);
  sm += __shfl_xor(sm, 4);
  sm += __shfl_xor(sm, 2);
  sm += __shfl_xor(sm, 1);
  const float aw_lane = ex / sm;            // valid in lanes 0..15

  // ---- per-lane distributed scalars ----
  const float off_lane = offs[qh * 32 + lane];                       // 32 values
  const size_t rbase = ((size_t)b * LQ + q) * (NL * 2);
  const float rp_lane = (lane < 8) ? refpts[rbase + lane] : 0.0f;    // 8 values

  const float* __restrict__ vb = value + (size_t)b * LIN * C_DIM;

  const int lsi[NL] = { 0, 16384, 20480, 21504 };

  float acc = 0.0f;
#pragma unroll
  for (int l = 0; l < NL; ++l) {
    const int   Hl = 128 >> l;
    const int   Wl = 128 >> l;
    const int   st = lsi[l];
    const float rx = __shfl(rp_lane, 2 * l);
    const float ry = __shfl(rp_lane, 2 * l + 1);
#pragma unroll
    for (int p = 0; p < NP; ++p) {
      const int lp = l * NP + p;
      const float ox = __shfl(off_lane, 2 * lp);
      const float oy = __shfl(off_lane, 2 * lp + 1);
      const float aw = __shfl(aw_lane, lp);

      // loc*size - 0.5  (grid_sample align_corners=False)
      const float x = fmaf(rx, (float)Wl, ox) - 0.5f;
      const float y = fmaf(ry, (float)Hl, oy) - 0.5f;
      const float x0f = floorf(x), y0f = floorf(y);
      const float lw = x - x0f, lh = y - y0f;
      const int x0 = (int)x0f, y0 = (int)y0f;

      const float w00 = (1.0f - lw) * (1.0f - lh);
      const float w10 = lw * (1.0f - lh);
      const float w01 = (1.0f - lw) * lh;
      const float w11 = lw * lh;

#pragma unroll
      for (int c = 0; c < 4; ++c) {
        const int ix = x0 + (c & 1);
        const int iy = y0 + (c >> 1);
        const float cw = (c == 0) ? w00 : (c == 1) ? w10 : (c == 2) ? w01 : w11;
        const bool valid = ((unsigned)ix < (unsigned)Wl) && ((unsigned)iy < (unsigned)Hl);
        int idx = iy * Wl + ix;
        idx = idx < 0 ? 0 : (idx > Hl * Wl - 1 ? Hl * Wl - 1 : idx);
        const float v = vb[((size_t)(st + idx) * NH + h) * HD + lane];
        acc = fmaf(aw * (valid ? cw : 0.0f), v, acc);
      }
    }
  }

  outpre[qh * HD + lane] = acc;
}

// =================================================================
// Host-side launcher
// =================================================================
static void launch_gemm(const float* A, const float* W, const float* bias,
                        float* Cm, int M, int N, int K, hipStream_t s) {
  dim3 grid((unsigned)((M / 16 + 7) / 8), (unsigned)(N / 64));
  hipLaunchKernelGGL(gemm_f32_wmma, grid, dim3(256), 0, s, A, W, bias, Cm, M, N, K);
}

extern "C" void kernel_launch(void* const* d_in, const int* in_sizes, int n_in,
                              void* d_out, int out_size, void* d_ws, size_t ws_size,
                              hipStream_t stream) {
  (void)in_sizes; (void)n_in; (void)out_size; (void)ws_size;

  const float* query   = (const float*)d_in[0];   // (B, LQ, C)
  const float* refpts  = (const float*)d_in[1];   // (B, LQ, NL, 2)
  const float* inflat  = (const float*)d_in[2];   // (B, LIN, C)
  const float* W_value = (const float*)d_in[3];
  const float* b_value = (const float*)d_in[4];
  const float* W_off   = (const float*)d_in[5];   // (C, 256)
  const float* b_off   = (const float*)d_in[6];
  const float* W_attn  = (const float*)d_in[7];   // (C, 128)
  const float* b_attn  = (const float*)d_in[8];
  const float* W_out   = (const float*)d_in[9];
  const float* b_out   = (const float*)d_in[10];

  float* out = (float*)d_out;

  const int M = B_SZ * LQ;                 // 43520 (== B*LIN)

  // workspace carve (floats)
  float* ws      = (float*)d_ws;
  float* val_ws  = ws;                                   // B*LIN*256
  float* off_ws  = val_ws + (size_t)B_SZ * LIN * 256;    // B*LQ*256
  float* attn_ws = off_ws + (size_t)B_SZ * LQ * 256;     // B*LQ*128
  float* pre_ws  = attn_ws + (size_t)B_SZ * LQ * 128;    // B*LQ*256

  // 1) value projection
  launch_gemm(inflat, W_value, b_value, val_ws, M, 256, C_DIM, stream);
  // 2) sampling-offset projection
  launch_gemm(query, W_off, b_off, off_ws, M, 256, C_DIM, stream);
  // 3) attention-logit projection
  launch_gemm(query, W_attn, b_attn, attn_ws, M, 128, C_DIM, stream);

  // 4) fused softmax + deformable bilinear sampling
  {
    const long long waves = (long long)B_SZ * LQ * NH;   // 348160
    const long long blocks = (waves * 32 + 255) / 256;   // 43520
    hipLaunchKernelGGL(msda_sample, dim3((unsigned)blocks), dim3(256), 0, stream,
                       val_ws, refpts, off_ws, attn_ws, pre_ws);
  }

  // 5) output projection -> d_out
  launch_gemm(pre_ws, W_out, b_out, out, M, 256, C_DIM, stream);
}